// VQNeighbor_26405458936343
// MI455X (gfx1250) — compile-verified
//
#include <hip/hip_runtime.h>
#include <hip/hip_bf16.h>

// ---------------- problem constants ----------------
#define BB 64
#define TT 512
#define ED 512
#define NE 1024          // n_e
#define NC 1025          // n_e + 1 codes
#define NPAD 1056        // 33 * 32, padded column count for D matrix
#define ROWS (BB * TT)   // 32768

// output layout (floats, concatenated in reference return order)
#define KH_OFF   0
#define ENC_OFF  (ROWS * ED)           // 16777216
#define V_OFF    (ENC_OFF + ROWS)      // 16809984
#define LOSS_OFF (V_OFF + 1)           // 16809985
#define KEM_OFF  (LOSS_OFF + 1)        // 16809986
#define EN_OFF   (KEM_OFF + ROWS)
#define EA_OFF   (EN_OFF + ROWS)

typedef __attribute__((ext_vector_type(2))) float v2f;
typedef __attribute__((ext_vector_type(8))) float v8f;

// ---------------- kernel 1: row squared norms (wave per row) ----------------
__global__ void __launch_bounds__(256)
sqnorm_rows(const float* __restrict__ X, float* __restrict__ out, int nvalid) {
    int wave = (int)((blockIdx.x * blockDim.x + threadIdx.x) >> 5);
    int lane = threadIdx.x & 31;
    if (wave >= nvalid) {                 // zero-fill padding rows (e2 tail)
        if (lane == 0) out[wave] = 0.0f;
        return;
    }
    const float4* row = (const float4*)(X + (size_t)wave * ED);
    float s = 0.0f;
#pragma unroll
    for (int i = 0; i < ED / 128; ++i) {  // 4 iters: 32 lanes * float4 = 128 floats
        float4 v = row[lane + i * 32];
        s += v.x * v.x + v.y * v.y + v.z * v.z + v.w * v.w;
    }
#pragma unroll
    for (int off = 16; off; off >>= 1) s += __shfl_xor(s, off);
    if (lane == 0) out[wave] = s;
}

// ---------------- kernel 2: distance GEMM via V_WMMA_F32_16X16X4_F32 -------
// Each wave computes a 32(M) x 32(N) tile of D = ks2 + e2 - 2 * KS @ EW^T
// using 2x2 register blocking: 4 independent 16x16 accumulators, so each
// K-step issues 4 loads -> 4 WMMAs (8 FLOP/byte from cache, no RAW chains).
// A[m][k]: lane%16 = m, (lane/16)*2 + j = k        (16x4 f32 A layout)
// B[k][n] = EW[n][k]: lane%16 = n, (lane/16)*2 + j = k
// C/D: VGPR r holds M = r + 8*(lane/16), N = lane%16
__global__ void __launch_bounds__(256)
dist_gemm(const float* __restrict__ KS, const float* __restrict__ EW,
          const float* __restrict__ ks2, const float* __restrict__ e2,
          float* __restrict__ Dm) {
    const int lane  = threadIdx.x & 31;
    const int wave  = threadIdx.x >> 5;                  // 0..7
    const int tileM = (blockIdx.x * 8 + wave) * 32;      // gridDim.x = ROWS/256
    const int tileN = blockIdx.y * 32;                   // gridDim.y = NPAD/32
    const int m     = lane & 15;
    const int khalf = (lane >> 4) * 2;                   // 0 or 2

    const float* a0p = KS + (size_t)(tileM + m) * ED + khalf;
    const float* a1p = a0p + (size_t)16 * ED;

    int n0 = tileN + m;
    int n1 = n0 + 16;
    int n0c = (n0 > NE) ? NE : n0;                       // pad cols read row NE (unused)
    int n1c = (n1 > NE) ? NE : n1;
    const float* b0p = EW + (size_t)n0c * ED + khalf;
    const float* b1p = EW + (size_t)n1c * ED + khalf;

    v8f acc00 = {}, acc01 = {}, acc10 = {}, acc11 = {};
#pragma unroll 2
    for (int k = 0; k < ED; k += 4) {
        v2f a0 = *(const v2f*)(a0p + k);
        v2f a1 = *(const v2f*)(a1p + k);
        v2f b0 = *(const v2f*)(b0p + k);
        v2f b1 = *(const v2f*)(b1p + k);
        acc00 = __builtin_amdgcn_wmma_f32_16x16x4_f32(false, a0, false, b0,
                                                      (short)0, acc00, false, false);
        acc01 = __builtin_amdgcn_wmma_f32_16x16x4_f32(false, a0, false, b1,
                                                      (short)0, acc01, false, false);
        acc10 = __builtin_amdgcn_wmma_f32_16x16x4_f32(false, a1, false, b0,
                                                      (short)0, acc10, false, false);
        acc11 = __builtin_amdgcn_wmma_f32_16x16x4_f32(false, a1, false, b1,
                                                      (short)0, acc11, false, false);
    }

    // epilogue: D = ks2[row] + e2[col] - 2*acc
    const int col0 = tileN + m;
    const int col1 = col0 + 16;
    const float e20 = e2[col0];
    const float e21 = e2[col1];
    const int mb = (lane >> 4) * 8;
#pragma unroll
    for (int r = 0; r < 8; ++r) {
        int row0 = tileM + mb + r;
        int row1 = row0 + 16;
        float k0 = ks2[row0];
        float k1 = ks2[row1];
        Dm[(size_t)row0 * NPAD + col0] = k0 + e20 - 2.0f * acc00[r];
        Dm[(size_t)row0 * NPAD + col1] = k0 + e21 - 2.0f * acc01[r];
        Dm[(size_t)row1 * NPAD + col0] = k1 + e20 - 2.0f * acc10[r];
        Dm[(size_t)row1 * NPAD + col1] = k1 + e21 - 2.0f * acc11[r];
    }
}

// ---------------- kernel 3: per-row argmin over 1025 codes ------------------
__global__ void __launch_bounds__(256)
row_argmin(const float* __restrict__ Dm, int* __restrict__ midx,
           float* __restrict__ mval) {
    int wave = (int)((blockIdx.x * blockDim.x + threadIdx.x) >> 5);
    int lane = threadIdx.x & 31;
    if (wave >= ROWS) return;
    const float* row = Dm + (size_t)wave * NPAD;
    float best = 3.4e38f;
    int bidx = 0;
    for (int c = lane; c < NC; c += 32) {
        float v = row[c];
        if (v < best) { best = v; bidx = c; }
    }
#pragma unroll
    for (int off = 16; off; off >>= 1) {
        float ov = __shfl_xor(best, off);
        int   oi = __shfl_xor(bidx, off);
        if (ov < best || (ov == best && oi < bidx)) { best = ov; bidx = oi; }
    }
    if (lane == 0) { midx[wave] = bidx; mval[wave] = best; }
}

// ---------------- kernel 4: sequential neighbor scan (thread per b) ---------
__global__ void scan_kernel(const float* __restrict__ Dm,
                            const int* __restrict__ minidx,
                            const float* __restrict__ minval,
                            const unsigned char* __restrict__ zmask,
                            int* __restrict__ encidx,
                            float* __restrict__ out,
                            float* __restrict__ partials,
                            int* __restrict__ ranges) {
    int b = threadIdx.x;
    if (b >= BB) return;
    const float eps = 1e-06f / (float)NE;

    int ind = minidx[(size_t)b * TT];
    if (ind > NE - 1) ind = NE - 1;
    if (zmask[b]) ind = 0;

    float prev_kem = 0.0f, lacc = 0.0f;
    int imax = ind, imin = ind;

    for (int t = 0; t < TT; ++t) {
        int bt = b * TT + t;
        size_t rowbase = (size_t)bt * NPAD;
        bool changed = false;
        if (t > 0) {
            float dh = Dm[rowbase + ind];
            int indn = (ind + 1 > NE - 1) ? NE - 1 : ind + 1;
            float dn = Dm[rowbase + indn];
            int ni = (dh <= dn) ? ind : indn;
            changed = (ni != ind);
            ind = ni;
        }
        float em_here = 1.2f * Dm[rowbase + ind];        // (1+LEGACY_ENERGY)*||ks-e||^2
        float em_next = 1.2f * Dm[rowbase + ind + 1];
        float kem = em_next - em_here;
        float lmin = 1.2f * minval[bt];                  // (1+LEGACY_CLUSTER)*d_min
        float reg_persist = __expf(-em_next);
        float e_norm = ((em_here > lmin - eps) ? (em_here - lmin + eps) : em_here)
                       + reg_persist;
        float e_abn = em_next + __expf(-em_here);

        encidx[bt]         = ind;
        out[ENC_OFF + bt]  = (float)ind;
        out[KEM_OFF + bt]  = kem;
        out[EN_OFF + bt]   = e_norm;
        out[EA_OFF + bt]   = e_abn;

        if (t > 0) {
            float ch = (kem - prev_kem) * (changed ? 0.0f : 1.0f) + eps;
            lacc += (ch > 0.0f) ? ch : 0.0f;
        }
        prev_kem = kem;
        if (ind > imax) imax = ind;
        if (ind < imin) imin = ind;
    }
    partials[b] = lacc;
    ranges[b]   = imax - imin;
}

// ---------------- kernel 5: final scalar reductions -------------------------
__global__ void finalize(const float* __restrict__ partials,
                         const int* __restrict__ ranges,
                         float* __restrict__ out) {
    __shared__ float sf[BB];
    __shared__ int   si[BB];
    int t = threadIdx.x;
    sf[t] = partials[t];
    si[t] = ranges[t];
    __syncthreads();
    for (int s = 32; s; s >>= 1) {
        if (t < s) {
            sf[t] += sf[t + s];
            if (si[t + s] > si[t]) si[t] = si[t + s];
        }
        __syncthreads();
    }
    if (t == 0) {
        out[LOSS_OFF] = sf[0] / (float)(BB * (TT - 1));
        out[V_OFF]    = (float)si[0];
    }
}

// ---------------- kernel 6: key_hard gather (straight-through forward) ------
__global__ void __launch_bounds__(128)
gather_hard(const float* __restrict__ EW, const int* __restrict__ encidx,
            float* __restrict__ out) {
    int bt  = blockIdx.x;
    int idx = encidx[bt];
    const float4* src = (const float4*)(EW + (size_t)idx * ED);
    float4* dst = (float4*)(out + KH_OFF + (size_t)bt * ED);
    dst[threadIdx.x] = src[threadIdx.x];   // 128 * float4 = 512 floats
}

// ---------------- launch ----------------------------------------------------
extern "C" void kernel_launch(void* const* d_in, const int* in_sizes, int n_in,
                              void* d_out, int out_size, void* d_ws, size_t ws_size,
                              hipStream_t stream) {
    const float*         key_soft = (const float*)d_in[0];
    const unsigned char* zmask    = (const unsigned char*)d_in[1];
    const float*         emb      = (const float*)d_in[2];
    float*               out      = (float*)d_out;

    // workspace carve-up
    float* Dm      = (float*)d_ws;                         // ROWS * NPAD
    float* ks2     = Dm + (size_t)ROWS * NPAD;             // ROWS
    float* e2      = ks2 + ROWS;                           // NPAD
    int*   minidx  = (int*)(e2 + NPAD);                    // ROWS
    float* mval    = (float*)(minidx + ROWS);              // ROWS
    int*   encidx  = (int*)(mval + ROWS);                  // ROWS
    float* partial = (float*)(encidx + ROWS);              // BB
    int*   ranges  = (int*)(partial + BB);                 // BB

    sqnorm_rows<<<ROWS / 8, 256, 0, stream>>>(key_soft, ks2, ROWS);
    sqnorm_rows<<<NPAD / 8, 256, 0, stream>>>(emb, e2, NC);

    dim3 ggrid(ROWS / 256, NPAD / 32);                     // (128, 33)
    dist_gemm<<<ggrid, 256, 0, stream>>>(key_soft, emb, ks2, e2, Dm);

    row_argmin<<<ROWS / 8, 256, 0, stream>>>(Dm, minidx, mval);

    scan_kernel<<<1, 64, 0, stream>>>(Dm, minidx, mval, zmask,
                                      encidx, out, partial, ranges);

    finalize<<<1, 64, 0, stream>>>(partial, ranges, out);

    gather_hard<<<ROWS, 128, 0, stream>>>(emb, encidx, out);
}